// peepLSTM_833223655468
// MI455X (gfx1250) — compile-verified
//
#include <hip/hip_runtime.h>
#include <math.h>

// Sizes from the reference
#define B_TOT 4096
#define S_LEN 512
#define E_DIM 64
#define H_DIM 256
#define C_DIM 10
#define M_TILE 32     // batch rows per workgroup (2 WMMA m-tiles)
#define KT 64         // 256 / 4 (K steps of the f32 WMMA)

typedef __attribute__((ext_vector_type(2))) float v2f;
typedef __attribute__((ext_vector_type(8))) float v8f;

__device__ __forceinline__ float sigmoidf_(float v) {
    return 1.0f / (1.0f + __expf(-v));
}

// ---------------------------------------------------------------------------
// Prologue 1: token lookup tables (VOCAB=3).
//   lutIF[v][0:256)   = emb[v] @ Wxi + bi      (i-gate preact)
//   lutIF[v][256:512) = emb[v] @ Wxf + bf      (f-gate preact)
//   lutG [v][h]       = sigmoid(emb[v] @ Wc + bc)   (candidate, no recurrence!)
//   lutO [v][h]       = emb[v] @ Wxo + bo      (o-gate preact, last step only)
// ---------------------------------------------------------------------------
__global__ __launch_bounds__(256) void build_luts(
    const float* __restrict__ emb,
    const float* __restrict__ Wxi, const float* __restrict__ bi,
    const float* __restrict__ Wxf, const float* __restrict__ bf,
    const float* __restrict__ Wxo, const float* __restrict__ bo,
    const float* __restrict__ Wc,  const float* __restrict__ bc,
    float* __restrict__ lutIF, float* __restrict__ lutG, float* __restrict__ lutO)
{
    int idx = blockIdx.x * blockDim.x + threadIdx.x;   // 0..3071
    if (idx >= 3072) return;
    const float* W; const float* b; float* dst; int v, col; bool sig = false;
    if (idx < 1536) {
        v = idx >> 9; int n = idx & 511;
        if (n < 256) { W = Wxi; b = bi; col = n; } else { W = Wxf; b = bf; col = n - 256; }
        dst = lutIF + idx;
    } else if (idx < 2304) {
        int r = idx - 1536; v = r >> 8; col = r & 255;
        W = Wc; b = bc; dst = lutG + r; sig = true;
    } else {
        int r = idx - 2304; v = r >> 8; col = r & 255;
        W = Wxo; b = bo; dst = lutO + r;
    }
    float s = b[col];
    #pragma unroll 8
    for (int e = 0; e < E_DIM; ++e) s += emb[v * E_DIM + e] * W[e * H_DIM + col];
    *dst = sig ? sigmoidf_(s) : s;
}

// ---------------------------------------------------------------------------
// Prologue 2: pre-swizzle recurrent weights into WMMA B-fragment order.
// packedIF: 32 n-tiles (0..15 = Wci cols, 16..31 = Wcf cols), packedO: 16 tiles.
// Element (nt, kt, lane, reg) holds W[k][n] with
//   k = 4*kt + 2*(lane>=16) + reg,  n = 16*nt + (lane&15)
// so the main loop's B fetch is one coalesced b64 per lane.
// ---------------------------------------------------------------------------
__global__ __launch_bounds__(256) void pack_weights(
    const float* __restrict__ Wci, const float* __restrict__ Wcf,
    const float* __restrict__ Wco,
    float* __restrict__ packedIF, float* __restrict__ packedO)
{
    int idx = blockIdx.x * blockDim.x + threadIdx.x;   // 0..196607
    if (idx >= 196608) return;
    int isO  = (idx >= 131072);
    int base = isO ? (idx - 131072) : idx;
    int nt   = base >> 12;        // 64*64 elems per n-tile
    int rem  = base & 4095;
    int kt   = rem >> 6;
    int li   = rem & 63;
    int lane = li >> 1;
    int reg  = li & 1;
    int k    = kt * 4 + ((lane >= 16) ? 2 : 0) + reg;
    int nlo  = lane & 15;
    if (!isO) {
        float v = (nt < 16) ? Wci[k * H_DIM + nt * 16 + nlo]
                            : Wcf[k * H_DIM + (nt - 16) * 16 + nlo];
        packedIF[base] = v;
    } else {
        packedO[base] = Wco[k * H_DIM + nt * 16 + nlo];
    }
}

// ---------------------------------------------------------------------------
// Main persistent-tile kernel: 128 workgroups x 512 threads (16 wave32).
// Carry c lives in LDS in A-fragment layout; wave w owns h-cols [16w,16w+16)
// for i, f (and o on the last step) so the gate merge is register-local.
// ---------------------------------------------------------------------------
__global__ __launch_bounds__(512) void lstm_main(
    const int* __restrict__ x,
    const float* __restrict__ packedIF, const float* __restrict__ packedO,
    const float* __restrict__ lutIF_g, const float* __restrict__ lutG_g,
    const float* __restrict__ lutO_g,
    const float* __restrict__ Wp, const float* __restrict__ bp,
    float* __restrict__ out)
{
    __shared__ float sA[2][KT][64];        // carry c, A-fragment layout (32 KB)
    __shared__ float sLutIF[3 * 512];
    __shared__ float sLutG[3 * 256];
    __shared__ float sLutO[3 * 256];
    __shared__ int   sTok[M_TILE];
    __shared__ float sP[M_TILE][C_DIM];

    const int tid     = threadIdx.x;
    const int lane    = tid & 31;
    const int wv      = tid >> 5;          // 0..15, owns h-cols [16wv,16wv+16)
    const int laneLo  = lane & 15;
    const int laneHi  = lane >> 4;
    const int baseRow = blockIdx.x * M_TILE;
    const int hcol    = wv * 16 + laneLo;  // this lane's h column

    for (int i = tid; i < 3 * 512; i += 512) sLutIF[i] = lutIF_g[i];
    for (int i = tid; i < 3 * 256; i += 512) { sLutG[i] = lutG_g[i]; sLutO[i] = lutO_g[i]; }
    for (int i = tid; i < 2 * KT * 64; i += 512) ((float*)sA)[i] = 0.0f;  // c0 = 0

    v8f acc[2][2];   // [m-tile][0 = i-gate, 1 = f-gate]
    v8f accO[2];     // o-gate, only touched on the last step

    // frag address of element (mloc, hcol) inside sA[mt]
    const int ktc     = hcol >> 2;
    const int sublane = ((hcol >> 1) & 1) * 16;
    const int regc    = hcol & 1;

    for (int t = 0; t < S_LEN; ++t) {
        __syncthreads();                                         // (A)
        if (tid < M_TILE) sTok[tid] = x[(size_t)(baseRow + tid) * S_LEN + t];
        __syncthreads();                                         // (B)

        // init accumulators with token-dependent preactivations (bias folded)
        #pragma unroll
        for (int mt = 0; mt < 2; ++mt) {
            #pragma unroll
            for (int v = 0; v < 8; ++v) {
                int m  = mt * 16 + v + 8 * laneHi;               // C layout: M = v + 8*laneHi
                int tk = sTok[m];
                acc[mt][0][v] = sLutIF[tk * 512 + hcol];
                acc[mt][1][v] = sLutIF[tk * 512 + 256 + hcol];
            }
        }

        // GEMM: c @ [Wci|Wcf] for this wave's columns, full f32 WMMA
        const float* bI = packedIF + (size_t)(wv)      * KT * 64;
        const float* bF = packedIF + (size_t)(16 + wv) * KT * 64;
        for (int kt = 0; kt < KT; ++kt) {
            v2f a0  = *(const v2f*)&sA[0][kt][lane * 2];
            v2f a1  = *(const v2f*)&sA[1][kt][lane * 2];
            v2f bi_ = *(const v2f*)&bI[kt * 64 + lane * 2];
            v2f bf_ = *(const v2f*)&bF[kt * 64 + lane * 2];
            acc[0][0] = __builtin_amdgcn_wmma_f32_16x16x4_f32(false, a0, false, bi_, (short)0, acc[0][0], false, false);
            acc[1][0] = __builtin_amdgcn_wmma_f32_16x16x4_f32(false, a1, false, bi_, (short)0, acc[1][0], false, false);
            acc[0][1] = __builtin_amdgcn_wmma_f32_16x16x4_f32(false, a0, false, bf_, (short)0, acc[0][1], false, false);
            acc[1][1] = __builtin_amdgcn_wmma_f32_16x16x4_f32(false, a1, false, bf_, (short)0, acc[1][1], false, false);
        }

        const bool last = (t == S_LEN - 1);
        if (last) {   // o-gate GEMM needed only once (h only consumed at t=S-1)
            #pragma unroll
            for (int mt = 0; mt < 2; ++mt)
                #pragma unroll
                for (int v = 0; v < 8; ++v) {
                    int m = mt * 16 + v + 8 * laneHi;
                    accO[mt][v] = sLutO[sTok[m] * 256 + hcol];
                }
            const float* bO = packedO + (size_t)wv * KT * 64;
            for (int kt = 0; kt < KT; ++kt) {
                v2f a0  = *(const v2f*)&sA[0][kt][lane * 2];
                v2f a1  = *(const v2f*)&sA[1][kt][lane * 2];
                v2f bo_ = *(const v2f*)&bO[kt * 64 + lane * 2];
                accO[0] = __builtin_amdgcn_wmma_f32_16x16x4_f32(false, a0, false, bo_, (short)0, accO[0], false, false);
                accO[1] = __builtin_amdgcn_wmma_f32_16x16x4_f32(false, a1, false, bo_, (short)0, accO[1], false, false);
            }
        }

        __syncthreads();                                         // (C) all sA reads done

        // gate merge: c' = sigmoid(g_pre)*sigmoid(i_pre)... note g already sigmoided in LUT
        #pragma unroll
        for (int mt = 0; mt < 2; ++mt) {
            #pragma unroll
            for (int v = 0; v < 8; ++v) {
                int mloc = v + 8 * laneHi;
                int m    = mt * 16 + mloc;
                float g    = sLutG[sTok[m] * 256 + hcol];
                float cold = sA[mt][ktc][(mloc + sublane) * 2 + regc];
                float cnew = g * sigmoidf_(acc[mt][0][v]) + cold * sigmoidf_(acc[mt][1][v]);
                float st   = cnew;
                if (last) st = tanhf(cnew) * sigmoidf_(accO[mt][v]);   // store h instead
                sA[mt][ktc][(mloc + sublane) * 2 + regc] = st;
            }
        }
    }

    __syncthreads();
    // final projection p = tanh(h @ Wp + bp); sA now holds h in frag layout
    if (tid < M_TILE * C_DIM) {
        int r = tid / C_DIM, cls = tid % C_DIM;
        int mt = r >> 4, mloc = r & 15;
        float s = bp[cls];
        for (int h = 0; h < H_DIM; ++h) {
            float hv = sA[mt][h >> 2][(mloc + ((h >> 1) & 1) * 16) * 2 + (h & 1)];
            s += hv * Wp[h * C_DIM + cls];
        }
        sP[r][cls] = tanhf(s);
    }
    __syncthreads();
    if (tid < M_TILE) {   // per-row log_softmax over 10 classes
        float mx = -1e30f;
        #pragma unroll
        for (int c = 0; c < C_DIM; ++c) mx = fmaxf(mx, sP[tid][c]);
        float sum = 0.0f;
        #pragma unroll
        for (int c = 0; c < C_DIM; ++c) sum += __expf(sP[tid][c] - mx);
        float l = mx + __logf(sum);
        #pragma unroll
        for (int c = 0; c < C_DIM; ++c)
            out[(size_t)(baseRow + tid) * C_DIM + c] = sP[tid][c] - l;
    }
}

// ---------------------------------------------------------------------------
// Workspace layout (floats): packedIF 131072 | packedO 65536 | lutIF 1536 |
// lutG 768 | lutO 768  -> ~780 KB total.
// ---------------------------------------------------------------------------
extern "C" void kernel_launch(void* const* d_in, const int* in_sizes, int n_in,
                              void* d_out, int out_size, void* d_ws, size_t ws_size,
                              hipStream_t stream) {
    (void)in_sizes; (void)n_in; (void)out_size; (void)ws_size;
    const int*   x   = (const int*)  d_in[0];
    const float* emb = (const float*)d_in[1];
    const float* Wxi = (const float*)d_in[2];
    const float* Wci = (const float*)d_in[3];
    const float* bi  = (const float*)d_in[4];
    const float* Wxf = (const float*)d_in[5];
    const float* Wcf = (const float*)d_in[6];
    const float* bf  = (const float*)d_in[7];
    const float* Wxo = (const float*)d_in[8];
    const float* Wco = (const float*)d_in[9];
    const float* bo  = (const float*)d_in[10];
    const float* Wc  = (const float*)d_in[11];
    const float* bc  = (const float*)d_in[12];
    const float* Wp  = (const float*)d_in[13];
    const float* bp  = (const float*)d_in[14];

    float* ws       = (float*)d_ws;
    float* packedIF = ws;                   // 131072
    float* packedO  = packedIF + 131072;    // 65536
    float* lutIF    = packedO  + 65536;     // 1536
    float* lutG     = lutIF    + 1536;      // 768
    float* lutO     = lutG     + 768;       // 768

    build_luts<<<dim3(12), dim3(256), 0, stream>>>(emb, Wxi, bi, Wxf, bf, Wxo, bo,
                                                   Wc, bc, lutIF, lutG, lutO);
    pack_weights<<<dim3(768), dim3(256), 0, stream>>>(Wci, Wcf, Wco, packedIF, packedO);
    lstm_main<<<dim3(B_TOT / M_TILE), dim3(512), 0, stream>>>(
        x, packedIF, packedO, lutIF, lutG, lutO, Wp, bp, (float*)d_out);
}